// GCN_Temp_16595753632119
// MI455X (gfx1250) — compile-verified
//
#include <hip/hip_runtime.h>

// ---------------------------------------------------------------------------
// NNConv GNN on gfx1250 (MI455X).  msg = XA @ [We_flat ; be_mat] with
// XA[e, a*in+i] = ea[e,a]*x_src[e,i] (+ raw block for bias rows) — avoids the
// 512 MB per-edge weight tensor.  Weight-bandwidth bound (~27 MB @ 23.3 TB/s),
// so fp32 V_WMMA_F32_16X16X4_F32 keeps reference precision with throughput to
// spare.  Wave computes one 16-wide N-tile across all MT M-tiles: B fragment
// loaded once per K-step, A fragments staged into distinct regs (clause-able
// global_load_b64s), then MT WMMAs back-to-back.  Hot loop is clamp-free
// (A zero-padded, tail peeled) with pure pointer stepping in address_space(1).
// ---------------------------------------------------------------------------

typedef __attribute__((ext_vector_type(2))) float v2f;
typedef __attribute__((ext_vector_type(8))) float v8f;

typedef __attribute__((address_space(1))) const float* gcfp;   // global const float*
typedef __attribute__((address_space(1))) const v2f*   gcv2p;  // global const v2f*

#define NN 64
#define EE 128
#define GG 4
#define D0 9
#define EDIM 4
#define H1 2000
#define H2 500
#define H3 100
#define F1 1000
#define F2 100
#define F3 50

static inline int ceil_div(int a, int b) { return (a + b - 1) / b; }
static inline int round4(int a) { return (a + 3) & ~3; }

// XA row layout (row length ldaP, zero padded):
//   cols [0, ED*in)       : ea[e,a] * H[src[e], i]   (a = c/in, i = c%in)
//   cols [ED*in, ED*in+in): H[src[e], i]             (bias rows)
//   cols [ED*in+in, ldaP) : 0
__global__ void build_xa_kernel(float* __restrict__ XA,
                                const float* __restrict__ H,
                                const int* __restrict__ src,
                                const float* __restrict__ ea,
                                int E, int ED, int in, int ldaP) {
  int idx = blockIdx.x * blockDim.x + threadIdx.x;
  if (idx >= E * ldaP) return;
  int c = idx % ldaP;
  int e = idx / ldaP;
  int K1 = ED * in;
  float v = 0.0f;
  if (c < K1) {
    int a = c / in, i = c % in;
    v = ea[e * ED + a] * H[src[e] * in + i];
  } else if (c < K1 + in) {
    v = H[src[e] * in + (c - K1)];
  }
  XA[idx] = v;
}

// dst[m, 0..Kp) = src[m, 0..K) zero-padded
__global__ void pad_copy_kernel(float* __restrict__ dst,
                                const float* __restrict__ srcM,
                                int M, int K, int Kp) {
  int idx = blockIdx.x * blockDim.x + threadIdx.x;
  if (idx >= M * Kp) return;
  int c = idx % Kp;
  int m = idx / Kp;
  dst[idx] = (c < K) ? srcM[m * K + c] : 0.0f;
}

// One K-segment.  aptr[] advances by 4 floats per step; segment-1 K is always
// a multiple of 4 in our calls so the pointers land exactly on the next
// segment's columns.  Main loop is clamp/branch-free; tail (Kseg % 4 != 0)
// clamps B rows — safe because A's pad columns are zero.
template <int MT>
__device__ inline void gemm_seg(v8f* acc, gcfp* aptr, gcfp B, int Kseg,
                                int N, int cnc, int half) {
  int kfull = Kseg & ~3;
  gcfp b0p = B + (size_t)(2 * half) * N + cnc;
  gcfp b1p = b0p + N;
  for (int k0 = 0; k0 < kfull; k0 += 4) {
    v2f b;
    b[0] = *b0p;
    b[1] = *b1p;
    v2f a[MT];                       // stage A frags in distinct registers
#pragma unroll
    for (int m = 0; m < MT; ++m) a[m] = *(gcv2p)aptr[m];
#pragma unroll
    for (int m = 0; m < MT; ++m) {
      acc[m] = __builtin_amdgcn_wmma_f32_16x16x4_f32(
          false, a[m], false, b, (short)0, acc[m], false, false);
      aptr[m] += 4;
    }
    b0p += (size_t)4 * N;
    b1p += (size_t)4 * N;
  }
  if (kfull < Kseg) {  // peeled tail
    int kr = kfull + 2 * half;
    int kb0 = min(kr, Kseg - 1);
    int kb1 = min(kr + 1, Kseg - 1);
    v2f b;
    b[0] = B[(size_t)kb0 * N + cnc];
    b[1] = B[(size_t)kb1 * N + cnc];
    v2f a[MT];
#pragma unroll
    for (int m = 0; m < MT; ++m) a[m] = *(gcv2p)aptr[m];
#pragma unroll
    for (int m = 0; m < MT; ++m) {
      acc[m] = __builtin_amdgcn_wmma_f32_16x16x4_f32(
          false, a[m], false, b, (short)0, acc[m], false, false);
      aptr[m] += 4;
    }
  }
}

// C[MT*16, N] = A[MT*16, *] @ [B1 rows 0..K1 ; B2 rows 0..K2]
// Wave handles one 16-wide N-tile across all MT row-tiles.
template <int MT>
__global__ void wmma_gemm_f32(const float* __restrict__ A, int lda,
                              const float* __restrict__ B1, int K1,
                              const float* __restrict__ B2, int K2,
                              float* __restrict__ C, int N) {
  int waveId = threadIdx.x / 32;
  int lane   = threadIdx.x % 32;
  int tilesN = (N + 15) / 16;
  int tn = blockIdx.x * (blockDim.x / 32) + waveId;
  if (tn >= tilesN) return;                  // wave-uniform exit
  int half = lane >> 4;
  int l16  = lane & 15;
  int cn   = tn * 16 + l16;
  int cnc  = min(cn, N - 1);                 // clamped load column

  v8f acc[MT];
  gcfp aptr[MT];
#pragma unroll
  for (int m = 0; m < MT; ++m) {
    acc[m] = (v8f){};
    aptr[m] = (gcfp)(uintptr_t)(A + (size_t)(m * 16 + l16) * lda + 2 * half);
  }

  gemm_seg<MT>(acc, aptr, (gcfp)(uintptr_t)B1, K1, N, cnc, half);
  if (K2 > 0) gemm_seg<MT>(acc, aptr, (gcfp)(uintptr_t)B2, K2, N, cnc, half);

  if (cn < N) {
#pragma unroll
    for (int m = 0; m < MT; ++m)
#pragma unroll
      for (int r = 0; r < 8; ++r)
        C[(size_t)(m * 16 + r + 8 * half) * N + cn] = acc[m][r];
  }
}

// Hout[n,o] = relu( rootOut[n,o] + bias[o] + sum_{e: dst[e]==n} msg[e,o] )
// Fixed edge order -> bitwise deterministic (no float atomics).
__global__ void combine_kernel(float* __restrict__ Hout,
                               const float* __restrict__ rootOut,
                               const float* __restrict__ msg,
                               const float* __restrict__ bias,
                               const int* __restrict__ dst,
                               int Nn, int E, int out) {
  int idx = blockIdx.x * blockDim.x + threadIdx.x;
  if (idx >= Nn * out) return;
  int o = idx % out;
  int n = idx / out;
  float s = rootOut[idx] + bias[o];
  for (int e = 0; e < E; ++e)
    if (dst[e] == n) s += msg[(size_t)e * out + o];
  Hout[idx] = fmaxf(s, 0.0f);
}

__global__ void readout_kernel(float* __restrict__ g,
                               const float* __restrict__ H,
                               const int* __restrict__ batch,
                               int Gn, int Nn, int F) {
  int idx = blockIdx.x * blockDim.x + threadIdx.x;
  if (idx >= Gn * F) return;
  int o = idx % F;
  int gi = idx / F;
  float s = 0.0f;
  for (int n = 0; n < Nn; ++n)
    if (batch[n] == gi) s += H[(size_t)n * F + o];
  g[idx] = s;
}

__global__ void dense_kernel(const float* __restrict__ X,
                             const float* __restrict__ W,
                             const float* __restrict__ bias,
                             float* __restrict__ Y,
                             int M, int K, int Nout, int doRelu) {
  int idx = blockIdx.x * blockDim.x + threadIdx.x;
  if (idx >= M * Nout) return;
  int o = idx % Nout;
  int m = idx / Nout;
  float s = bias[o];
  for (int k = 0; k < K; ++k) s += X[(size_t)m * K + k] * W[(size_t)k * Nout + o];
  Y[idx] = doRelu ? fmaxf(s, 0.0f) : s;
}

extern "C" void kernel_launch(void* const* d_in, const int* in_sizes, int n_in,
                              void* d_out, int out_size, void* d_ws, size_t ws_size,
                              hipStream_t stream) {
  (void)in_sizes; (void)n_in; (void)out_size; (void)ws_size;

  const float* x     = (const float*)d_in[0];
  const int*   eidx  = (const int*)d_in[1];
  const float* ea    = (const float*)d_in[2];
  const int*   batch = (const int*)d_in[3];
  const float* We1 = (const float*)d_in[4];  const float* be1 = (const float*)d_in[5];
  const float* rt1 = (const float*)d_in[6];  const float* b1  = (const float*)d_in[7];
  const float* We2 = (const float*)d_in[8];  const float* be2 = (const float*)d_in[9];
  const float* rt2 = (const float*)d_in[10]; const float* b2  = (const float*)d_in[11];
  const float* We3 = (const float*)d_in[12]; const float* be3 = (const float*)d_in[13];
  const float* rt3 = (const float*)d_in[14]; const float* b3  = (const float*)d_in[15];
  const float* W1  = (const float*)d_in[16]; const float* c1  = (const float*)d_in[17];
  const float* W2  = (const float*)d_in[18]; const float* c2  = (const float*)d_in[19];
  const float* W3  = (const float*)d_in[20]; const float* c3  = (const float*)d_in[21];
  const float* W4  = (const float*)d_in[22]; const float* c4  = (const float*)d_in[23];

  const int* src = eidx;        // edge_index[0,:]
  const int* dst = eidx + EE;   // edge_index[1,:]

  // Workspace (floats).  All offsets even -> 8B alignment for v2f loads.
  float* ws = (float*)d_ws;
  size_t off = 0;
  float* XA      = ws + off; off += (size_t)EE * (EDIM + 1) * H1;  // 1,280,000
  float* msg     = ws + off; off += (size_t)EE * H1;               //   256,000
  float* rootOut = ws + off; off += (size_t)NN * H1;               //   128,000
  float* h1      = ws + off; off += (size_t)NN * H1;               //   128,000
  float* h2      = ws + off; off += (size_t)NN * H2;               //    32,000
  float* h3      = ws + off; off += (size_t)NN * H3;               //     6,400
  float* g0      = ws + off; off += (size_t)GG * H3;               //       400
  float* f1      = ws + off; off += (size_t)GG * F1;               //     4,000
  float* f2      = ws + off; off += (size_t)GG * F2;               //       400
  float* f3      = ws + off; off += (size_t)GG * F3;               //       200
  float* xP      = ws + off; off += (size_t)NN * round4(D0);       //       768

  auto gemm8 = [&](const float* A, int lda, const float* B1s, int K1,
                   const float* B2s, int K2, float* C, int Ncols) {
    int blocks = ceil_div(ceil_div(Ncols, 16), 8);
    wmma_gemm_f32<8><<<dim3(blocks), dim3(256), 0, stream>>>(
        A, lda, B1s, K1, B2s, K2, C, Ncols);
  };
  auto gemm4 = [&](const float* A, int lda, const float* B1s, int K1,
                   float* C, int Ncols) {
    int blocks = ceil_div(ceil_div(Ncols, 16), 8);
    wmma_gemm_f32<4><<<dim3(blocks), dim3(256), 0, stream>>>(
        A, lda, B1s, K1, (const float*)nullptr, 0, C, Ncols);
  };

  // x -> padded copy (K=9 -> 12) for the layer-1 root GEMM
  pad_copy_kernel<<<dim3(ceil_div(NN * round4(D0), 256)), dim3(256), 0, stream>>>(
      xP, x, NN, D0, round4(D0));

  // ---------------- Layer 1: in=9, out=2000 ----------------
  {
    int in = D0, out = H1;
    int K1 = EDIM * in, K2 = in;
    int ldaP = round4(K1 + K2);  // 48
    build_xa_kernel<<<dim3(ceil_div(EE * ldaP, 256)), dim3(256), 0, stream>>>(
        XA, x, src, ea, EE, EDIM, in, ldaP);
    gemm8(XA, ldaP, We1, K1, be1, K2, msg, out);
    gemm4(xP, round4(in), rt1, in, rootOut, out);
    combine_kernel<<<dim3(ceil_div(NN * out, 256)), dim3(256), 0, stream>>>(
        h1, rootOut, msg, b1, dst, NN, EE, out);
  }

  // ---------------- Layer 2: in=2000, out=500 ----------------
  {
    int in = H1, out = H2;
    int K1 = EDIM * in, K2 = in;
    int ldaP = K1 + K2;          // 10000 (mult of 4)
    build_xa_kernel<<<dim3(ceil_div(EE * ldaP, 256)), dim3(256), 0, stream>>>(
        XA, h1, src, ea, EE, EDIM, in, ldaP);
    gemm8(XA, ldaP, We2, K1, be2, K2, msg, out);
    gemm4(h1, in, rt2, in, rootOut, out);
    combine_kernel<<<dim3(ceil_div(NN * out, 256)), dim3(256), 0, stream>>>(
        h2, rootOut, msg, b2, dst, NN, EE, out);
  }

  // ---------------- Layer 3: in=500, out=100 ----------------
  {
    int in = H2, out = H3;
    int K1 = EDIM * in, K2 = in;
    int ldaP = K1 + K2;          // 2500
    build_xa_kernel<<<dim3(ceil_div(EE * ldaP, 256)), dim3(256), 0, stream>>>(
        XA, h2, src, ea, EE, EDIM, in, ldaP);
    gemm8(XA, ldaP, We3, K1, be3, K2, msg, out);
    gemm4(h2, in, rt3, in, rootOut, out);
    combine_kernel<<<dim3(ceil_div(NN * out, 256)), dim3(256), 0, stream>>>(
        h3, rootOut, msg, b3, dst, NN, EE, out);
  }

  // ---------------- Readout + MLP head ----------------
  readout_kernel<<<dim3(ceil_div(GG * H3, 256)), dim3(256), 0, stream>>>(
      g0, h3, batch, GG, NN, H3);
  dense_kernel<<<dim3(ceil_div(GG * F1, 256)), dim3(256), 0, stream>>>(
      g0, W1, c1, f1, GG, H3, F1, 1);
  dense_kernel<<<dim3(ceil_div(GG * F2, 256)), dim3(256), 0, stream>>>(
      f1, W2, c2, f2, GG, F1, F2, 1);
  dense_kernel<<<dim3(ceil_div(GG * F3, 256)), dim3(256), 0, stream>>>(
      f2, W3, c3, f3, GG, F2, F3, 1);
  dense_kernel<<<dim3(1), dim3(256), 0, stream>>>(
      f3, W4, c4, (float*)d_out, GG, F3, 1, 0);
}